// ESN_45578192945261
// MI455X (gfx1250) — compile-verified
//
#include <hip/hip_runtime.h>
#include <hip/hip_bf16.h>
#include <cstdint>

// ESN update: per-chunk sparse COO SpMM over 64 sequence vectors, then
// leaky tanh blend. Memory/atomic-bound (~80MB traffic -> ~3.5us @ 23.3TB/s).
// Strategy: transpose state so seq is innermost (coalesced gathers/atomics),
// stream COO tiles to LDS via gfx1250 async-to-LDS (confirmed:
// global_load_async_to_lds_b32 in s_clause), scatter with native f32 global
// atomics (accumulator fits in 192MB L2), fused tanh finalize.

constexpr int RES_DIM = 4096;
constexpr int CHUNKS  = 8;
constexpr int SEQ     = 64;
constexpr int NNZ     = 335544;            // int(0.02 * 4096 * 4096)
constexpr float LEAK  = 0.6f;
constexpr float BIAS  = 1.6f;
constexpr int NNZ_PER_BLOCK = 1024;

#define GLOBAL_AS __attribute__((address_space(1)))
#define LDS_AS    __attribute__((address_space(3)))

// ---- CDNA5 feature probes (guarded so compile never hard-fails) ----------
#if defined(__has_builtin)
#if __has_builtin(__builtin_amdgcn_global_load_async_to_lds_b32)
#define HAVE_ASYNC_LDS 1
#endif
#if __has_builtin(__builtin_amdgcn_s_wait_asynccnt)
#define HAVE_WAIT_ASYNC 1
#endif
#if __has_builtin(__builtin_amdgcn_tanhf)
#define HAVE_HW_TANH 1
#endif
#if __has_builtin(__builtin_amdgcn_global_atomic_fadd_f32)
#define HAVE_GLOBAL_FADD 1
#endif
#endif

__device__ __forceinline__ void atomic_add_f32(float* p, float v) {
#if defined(HAVE_GLOBAL_FADD)
    __builtin_amdgcn_global_atomic_fadd_f32((GLOBAL_AS float*)p, v);
#else
    // relaxed agent-scope fadd: should lower to global_atomic_add_f32 on gfx1250
    __hip_atomic_fetch_add(p, v, __ATOMIC_RELAXED, __HIP_MEMORY_SCOPE_AGENT);
#endif
}

__device__ __forceinline__ float tanh_f32(float x) {
#if defined(HAVE_HW_TANH)
    return __builtin_amdgcn_tanhf(x);   // gfx1250 v_tanh_f32
#else
    return tanhf(x);
#endif
}

// --------------------------------------------------------------------------
// COMPILE PROBE (never launched): placed first in the file so the disasm
// snippet shows exactly how the f32 atomic add and tanh lower on gfx1250.
__global__ void esn_codegen_probe(float* __restrict__ p,
                                  const float* __restrict__ q) {
    atomic_add_f32(&p[threadIdx.x], q[threadIdx.x]);
    p[threadIdx.x + 1024] = tanh_f32(q[threadIdx.x + 1024]);
}

// --------------------------------------------------------------------------
// Kernel: COO scatter. Each block stages 1024 (val,row,col) triples into LDS
// as packed 16B records via gfx1250 async-to-LDS, then each wave takes one
// nonzero per step (single ds_load_b128 broadcast): lanes cover s=0..31 and
// s=32..63 -> coalesced gather from XT (L2-hot) and coalesced
// global_atomic_add_f32 into wxT (L2-resident, 8MB total).
// grid = (ceil(NNZ/1024), CHUNKS), block = 256 (8 waves)
__global__ __launch_bounds__(256) void esn_scatter(const float* __restrict__ vals,
                                                   const int* __restrict__ rows,
                                                   const int* __restrict__ cols,
                                                   const float* __restrict__ XT,
                                                   float* __restrict__ wxT) {
    __shared__ int4 s_coo[NNZ_PER_BLOCK];   // {val_bits, row, col, pad} -> 16KB

    const int c    = blockIdx.y;
    const int base = blockIdx.x * NNZ_PER_BLOCK;
    const int count = (NNZ - base < NNZ_PER_BLOCK) ? (NNZ - base) : NNZ_PER_BLOCK;

    const float* __restrict__ v  = vals + (size_t)c * NNZ;
    const int*   __restrict__ r  = rows + (size_t)c * NNZ;
    const int*   __restrict__ cl = cols + (size_t)c * NNZ;
    const float* __restrict__ X  = XT  + (size_t)c * RES_DIM * SEQ;
    float*       __restrict__ W  = wxT + (size_t)c * RES_DIM * SEQ;

    // --- stage COO tile into LDS ---
#if defined(HAVE_ASYNC_LDS)
    for (int i = threadIdx.x; i < count; i += 256) {
        const int n = base + i;
        __builtin_amdgcn_global_load_async_to_lds_b32(
            (GLOBAL_AS int*)(v + n),  (LDS_AS int*)(&s_coo[i].x), 0, 0);
        __builtin_amdgcn_global_load_async_to_lds_b32(
            (GLOBAL_AS int*)(r + n),  (LDS_AS int*)(&s_coo[i].y), 0, 0);
        __builtin_amdgcn_global_load_async_to_lds_b32(
            (GLOBAL_AS int*)(cl + n), (LDS_AS int*)(&s_coo[i].z), 0, 0);
    }
#if defined(HAVE_WAIT_ASYNC)
    __builtin_amdgcn_s_wait_asynccnt(0);
#else
    asm volatile("s_wait_asynccnt 0" ::: "memory");
#endif
    __syncthreads();
#else
    for (int i = threadIdx.x; i < count; i += 256) {
        const int n = base + i;
        s_coo[i] = make_int4(__float_as_int(v[n]), r[n], cl[n], 0);
    }
    __syncthreads();
#endif

    // --- scatter: one nonzero per wave per iteration ---
    const int wave = threadIdx.x >> 5;   // 0..7
    const int lane = threadIdx.x & 31;   // 0..31

    for (int i = wave; i < count; i += 8) {
        const int4 t = s_coo[i];         // uniform per wave -> ds_load_b128 bcast
        const float val = __int_as_float(t.x);
        const int   row = t.y;
        const int   col = t.z;
        const float* xs = X + (size_t)col * SEQ;
        float*       wd = W + (size_t)row * SEQ;
        const float x0 = xs[lane];
        const float x1 = xs[lane + 32];
        atomic_add_f32(&wd[lane],      val * x0);
        atomic_add_f32(&wd[lane + 32], val * x1);
    }
}

// --------------------------------------------------------------------------
// Kernel: XT[c][d][s] = res_state[s][c][d]  (seq innermost), and wxT = 0.
// 32x32 tiles through LDS; all global accesses coalesced.
// grid = (RES_DIM/32, SEQ/32, CHUNKS), block = (32, 8)
__global__ __launch_bounds__(256) void esn_prep(const float* __restrict__ res,
                                                float* __restrict__ XT,
                                                float* __restrict__ wxT) {
    __shared__ float tile[32][33];
    const int c  = blockIdx.z;
    const int d0 = blockIdx.x * 32;
    const int s0 = blockIdx.y * 32;
    const int lx = threadIdx.x;   // 0..31
    const int ly = threadIdx.y;   // 0..7

#pragma unroll
    for (int i = 0; i < 4; ++i) {
        const int s = s0 + ly + 8 * i;
        tile[ly + 8 * i][lx] = res[((size_t)s * CHUNKS + c) * RES_DIM + d0 + lx];
    }
    __syncthreads();
#pragma unroll
    for (int i = 0; i < 4; ++i) {
        const int d = d0 + ly + 8 * i;
        const size_t o = ((size_t)c * RES_DIM + d) * SEQ + s0 + lx;
        XT[o]  = tile[lx][ly + 8 * i];
        wxT[o] = 0.0f;
    }
}

// --------------------------------------------------------------------------
// Kernel: out[s][c][d] = LEAK*tanh(wxT[c][d][s] + proj[s][c][d] + BIAS)
//                      + (1-LEAK)*res[s][c][d]
// LDS 32x32 tile transpose so both sides stay coalesced.
// grid = (RES_DIM/32, SEQ/32, CHUNKS), block = (32, 8)
__global__ __launch_bounds__(256) void esn_finalize(const float* __restrict__ wxT,
                                                    const float* __restrict__ proj,
                                                    const float* __restrict__ res,
                                                    float* __restrict__ out) {
    __shared__ float tile[32][33];
    const int c  = blockIdx.z;
    const int d0 = blockIdx.x * 32;
    const int s0 = blockIdx.y * 32;
    const int lx = threadIdx.x;
    const int ly = threadIdx.y;

#pragma unroll
    for (int i = 0; i < 4; ++i) {
        const int d = d0 + ly + 8 * i;
        tile[ly + 8 * i][lx] = wxT[((size_t)c * RES_DIM + d) * SEQ + s0 + lx];
    }
    __syncthreads();
#pragma unroll
    for (int i = 0; i < 4; ++i) {
        const int s = s0 + ly + 8 * i;
        const int d = d0 + lx;
        const size_t o = ((size_t)s * CHUNKS + c) * RES_DIM + d;
        const float wx = tile[lx][ly + 8 * i];
        const float a  = tanh_f32(wx + proj[o] + BIAS);
        out[o] = LEAK * a + (1.0f - LEAK) * res[o];
    }
}

// --------------------------------------------------------------------------
extern "C" void kernel_launch(void* const* d_in, const int* in_sizes, int n_in,
                              void* d_out, int out_size, void* d_ws, size_t ws_size,
                              hipStream_t stream) {
    const float* proj = (const float*)d_in[0];   // (SEQ, CHUNKS, RES_DIM)
    const float* res  = (const float*)d_in[1];   // (SEQ, CHUNKS, RES_DIM)
    const float* vals = (const float*)d_in[2];   // (CHUNKS, NNZ)
    const int*   rows = (const int*)d_in[3];     // (CHUNKS, NNZ)
    const int*   cols = (const int*)d_in[4];     // (CHUNKS, NNZ)
    float* out = (float*)d_out;

    // workspace: XT (8MB) | wxT (8MB)  -> needs 16MB of d_ws
    float* XT  = (float*)d_ws;
    float* wxT = XT + (size_t)CHUNKS * RES_DIM * SEQ;

    const dim3 tb(32, 8);
    const dim3 tg(RES_DIM / 32, SEQ / 32, CHUNKS);
    esn_prep<<<tg, tb, 0, stream>>>(res, XT, wxT);

    const int bpc = (NNZ + NNZ_PER_BLOCK - 1) / NNZ_PER_BLOCK;
    esn_scatter<<<dim3(bpc, CHUNKS), 256, 0, stream>>>(vals, rows, cols, XT, wxT);

    esn_finalize<<<tg, tb, 0, stream>>>(wxT, proj, res, out);
}